// GCN_73959336837366
// MI455X (gfx1250) — compile-verified
//
#include <hip/hip_runtime.h>

#define NN    50000
#define NE    1600000
#define NG    512
#define INC   16
#define H1C   64
#define H2C   64
#define H3C   128
#define OUTC  10

typedef __attribute__((ext_vector_type(2))) float v2f;
typedef __attribute__((ext_vector_type(8))) float v8f;

// ---------------- utility ----------------
__global__ void fill_kernel(float* __restrict__ p, float v, int n) {
    int i = blockIdx.x * blockDim.x + threadIdx.x;
    if (i < n) p[i] = v;
}

__global__ void deg_kernel(const int* __restrict__ dst, float* __restrict__ deg, int E) {
    int e = blockIdx.x * blockDim.x + threadIdx.x;
    if (e < E) atomicAdd(&deg[dst[e]], 1.0f);
}

__global__ void rsqrt_kernel(float* __restrict__ deg, int n) {
    int i = blockIdx.x * blockDim.x + threadIdx.x;
    if (i < n) deg[i] = rsqrtf(deg[i]);
}

// ---------------- dense transform: C[M,NCOL] = A[M,K] @ B[K,NCOL] ----------------
// One wave32 per 16x16 output tile, V_WMMA_F32_16X16X4_F32 chained over K.
template<int K, int NCOL>
__global__ void wmma_gemm_f32(const float* __restrict__ A, const float* __restrict__ B,
                              float* __restrict__ C, int M) {
    const int wavesPerBlock = blockDim.x >> 5;
    int wave = blockIdx.x * wavesPerBlock + (threadIdx.x >> 5);
    const int colTiles = NCOL / 16;
    int totalTiles = (M / 16) * colTiles;
    if (wave >= totalTiles) return;          // uniform per-wave: EXEC stays all-1s

    int rowBase = (wave / colTiles) * 16;
    int colBase = (wave % colTiles) * 16;
    int lane = threadIdx.x & 31;
    int hi   = lane >> 4;                    // 0: lanes 0-15, 1: lanes 16-31
    int l    = lane & 15;

    v8f acc = {};
    #pragma unroll
    for (int kt = 0; kt < K; kt += 4) {
        int ka = kt + hi * 2;                // K pair held by this half-wave
        // A 16x4 f32 layout: lane=M, {VGPR0,VGPR1} = {K=ka, K=ka+1}; contiguous -> v2f load
        v2f a = *reinterpret_cast<const v2f*>(A + (size_t)(rowBase + l) * K + ka);
        // B 4x16 f32 layout: lane=N, {VGPR0,VGPR1} = rows {ka, ka+1}
        v2f b;
        b.x = B[(size_t)ka * NCOL + colBase + l];
        b.y = B[(size_t)(ka + 1) * NCOL + colBase + l];
        acc = __builtin_amdgcn_wmma_f32_16x16x4_f32(false, a, false, b,
                                                    (short)0, acc, false, false);
    }
    // C/D layout: VGPR v holds M = v + 8*hi, N = l
    #pragma unroll
    for (int v = 0; v < 8; ++v) {
        int m = rowBase + v + 8 * hi;
        C[(size_t)m * NCOL + colBase + l] = acc[v];
    }
}

// ---------------- edge scatter: agg[dst] += hw[src] * dinv[src]*dinv[dst] ----------------
// thread = (edge, 4-channel chunk); float4 gather, 4 f32 atomics (L2-resident targets)
__global__ void scatter_kernel(const int* __restrict__ src, const int* __restrict__ dst,
                               const float* __restrict__ dinv, const float* __restrict__ hw,
                               float* __restrict__ agg, int E) {
    int t = blockIdx.x * blockDim.x + threadIdx.x;
    int e = t >> 4;
    if (e >= E) return;
    int c = (t & 15) * 4;
    int s = src[e], d = dst[e];
    float w = dinv[s] * dinv[d];
    const float4 v = *reinterpret_cast<const float4*>(hw + (size_t)s * 64 + c);
    float* o = agg + (size_t)d * 64 + c;
    atomicAdd(o + 0, v.x * w);
    atomicAdd(o + 1, v.y * w);
    atomicAdd(o + 2, v.z * w);
    atomicAdd(o + 3, v.w * w);
}

// h = relu(agg + dinv^2 * hw + b)   (self-loop folded in analytically)
__global__ void bias_relu_kernel(const float* __restrict__ agg, const float* __restrict__ hw,
                                 const float* __restrict__ dinv, const float* __restrict__ b,
                                 float* __restrict__ out, int N) {
    int t = blockIdx.x * blockDim.x + threadIdx.x;
    if (t >= N * 64) return;
    int i = t >> 6, c = t & 63;
    float di = dinv[i];
    float v = agg[t] + di * di * hw[t] + b[c];
    out[t] = fmaxf(v, 0.0f);
}

// layer-2 epilogue fused with mean-pool accumulation: h2 never materialized
__global__ void bias_relu_pool_kernel(const float* __restrict__ agg, const float* __restrict__ hw,
                                      const float* __restrict__ dinv, const float* __restrict__ b,
                                      const int* __restrict__ batch, float* __restrict__ pooled,
                                      int N) {
    int t = blockIdx.x * blockDim.x + threadIdx.x;
    if (t >= N * 64) return;
    int i = t >> 6, c = t & 63;
    float di = dinv[i];
    float v = fmaxf(agg[t] + di * di * hw[t] + b[c], 0.0f);
    atomicAdd(&pooled[(size_t)batch[i] * 64 + c], v);
}

__global__ void count_kernel(const int* __restrict__ batch, float* __restrict__ counts, int N) {
    int i = blockIdx.x * blockDim.x + threadIdx.x;
    if (i < N) atomicAdd(&counts[batch[i]], 1.0f);
}

// ---------------- per-graph FC head: out = (pooled/cnt) @ Wf1 + bf1) @ Wf2 + bf2 ----------------
__global__ void fc_kernel(const float* __restrict__ pooled, const float* __restrict__ counts,
                          const float* __restrict__ Wf1, const float* __restrict__ bf1,
                          const float* __restrict__ Wf2, const float* __restrict__ bf2,
                          float* __restrict__ out) {
    __shared__ float pm[H2C];
    __shared__ float hid[H3C];
    int g = blockIdx.x;
    int t = threadIdx.x;                 // 128 threads
    float inv = 1.0f / fmaxf(counts[g], 1.0f);
    if (t < H2C) pm[t] = pooled[(size_t)g * H2C + t] * inv;
    __syncthreads();
    float acc = bf1[t];
    #pragma unroll 8
    for (int k = 0; k < H2C; ++k) acc += pm[k] * Wf1[k * H3C + t];
    hid[t] = acc;
    __syncthreads();
    if (t < OUTC) {
        float o = bf2[t];
        #pragma unroll 8
        for (int j = 0; j < H3C; ++j) o += hid[j] * Wf2[j * OUTC + t];
        out[(size_t)g * OUTC + t] = o;
    }
}

extern "C" void kernel_launch(void* const* d_in, const int* in_sizes, int n_in,
                              void* d_out, int out_size, void* d_ws, size_t ws_size,
                              hipStream_t stream) {
    const float* x    = (const float*)d_in[0];
    const int*   ei   = (const int*)d_in[1];
    const int*   src  = ei;              // edge_index[0]
    const int*   dst  = ei + NE;         // edge_index[1]
    const int*   batch= (const int*)d_in[2];
    const float* W1   = (const float*)d_in[3];
    const float* b1   = (const float*)d_in[4];
    const float* W2   = (const float*)d_in[5];
    const float* b2   = (const float*)d_in[6];
    const float* Wf1  = (const float*)d_in[7];
    const float* bf1  = (const float*)d_in[8];
    const float* Wf2  = (const float*)d_in[9];
    const float* bf2  = (const float*)d_in[10];
    float* out = (float*)d_out;

    // workspace layout (floats)
    float* ws     = (float*)d_ws;
    float* deg    = ws;                         // [NN]       degree -> dinv (in place)
    float* hw     = deg + NN;                   // [NN*64]    x@W1, then h1@W2 (reused)
    float* h1     = hw + (size_t)NN * 64;       // [NN*64]
    float* agg    = h1 + (size_t)NN * 64;       // [NN*64]    scatter accumulator (reused)
    float* pooled = agg + (size_t)NN * 64;      // [NG*64]
    float* counts = pooled + (size_t)NG * 64;   // [NG]

    const int BT = 256;
    auto blk = [](long long n, int b) { return (int)((n + b - 1) / b); };

    // degree with self-loop baked in as init=1.0
    fill_kernel<<<blk(NN, BT), BT, 0, stream>>>(deg, 1.0f, NN);
    fill_kernel<<<blk((size_t)NN * 64, BT), BT, 0, stream>>>(agg, 0.0f, NN * 64);
    fill_kernel<<<blk(NG * 64 + NG, BT), BT, 0, stream>>>(pooled, 0.0f, NG * 64 + NG);

    deg_kernel<<<blk(NE, BT), BT, 0, stream>>>(dst, deg, NE);
    rsqrt_kernel<<<blk(NN, BT), BT, 0, stream>>>(deg, NN);   // deg -> dinv

    // ---- layer 1 ----
    {
        int tiles = (NN / 16) * (H1C / 16);                  // 3125 * 4
        wmma_gemm_f32<INC, H1C><<<blk(tiles, 8), 256, 0, stream>>>(x, W1, hw, NN);
    }
    scatter_kernel<<<blk((long long)NE * 16, BT), BT, 0, stream>>>(src, dst, deg, hw, agg, NE);
    bias_relu_kernel<<<blk((size_t)NN * 64, BT), BT, 0, stream>>>(agg, hw, deg, b1, h1, NN);

    // ---- layer 2 ----
    fill_kernel<<<blk((size_t)NN * 64, BT), BT, 0, stream>>>(agg, 0.0f, NN * 64);
    {
        int tiles = (NN / 16) * (H2C / 16);
        wmma_gemm_f32<H1C, H2C><<<blk(tiles, 8), 256, 0, stream>>>(h1, W2, hw, NN);
    }
    scatter_kernel<<<blk((long long)NE * 16, BT), BT, 0, stream>>>(src, dst, deg, hw, agg, NE);
    bias_relu_pool_kernel<<<blk((size_t)NN * 64, BT), BT, 0, stream>>>(agg, hw, deg, b2, batch,
                                                                      pooled, NN);
    count_kernel<<<blk(NN, BT), BT, 0, stream>>>(batch, counts, NN);

    // ---- FC head ----
    fc_kernel<<<NG, H3C, 0, stream>>>(pooled, counts, Wf1, bf1, Wf2, bf2, out);
}